// DND_13065290514794
// MI455X (gfx1250) — compile-verified
//
#include <hip/hip_runtime.h>
#include <hip/hip_bf16.h>
#include <math.h>

// Problem dims (match reference)
#define LL 2048   // episode length
#define BB 64     // batch
#define KK 256    // key size
#define VV 256    // value size
#define HH 8      // heads
#define HK 2048   // H*K
#define HV 2048   // H*V

typedef __attribute__((ext_vector_type(2))) float v2f;
typedef __attribute__((ext_vector_type(8))) float v8f;

// D = A(16x4 f32) * B(4x16 f32) + C(16x16 f32)  -- V_WMMA_F32_16X16X4_F32
__device__ __forceinline__ v8f wmma4(v2f a, v2f b, v8f c) {
    return __builtin_amdgcn_wmma_f32_16x16x4_f32(
        /*neg_a=*/false, a, /*neg_b=*/false, b,
        /*c_mod=*/(short)0, c, /*reuse_a=*/false, /*reuse_b=*/false);
}

// ---------------------------------------------------------------------------
// Kernel 1: q = query[B,K] @ Wq[K,HK] + bq  ->  qbuf[B][HK]
// tiles: tm in [0,4), tn in [0,128); 512 waves
// ---------------------------------------------------------------------------
__global__ void k_qproj(const float* __restrict__ query,
                        const float* __restrict__ Wq,
                        const float* __restrict__ bq,
                        float* __restrict__ qbuf) {
    const int waveId = blockIdx.x * (blockDim.x >> 5) + (threadIdx.x >> 5);
    const int tm   = waveId >> 7;
    const int tn   = waveId & 127;
    const int lane = threadIdx.x & 31;
    const int col  = lane & 15;   // A row m / B,C,D column n
    const int half = lane >> 4;

    const float* Arow = query + (size_t)(tm * 16 + col) * KK;
    const float* Bcol = Wq + (tn * 16 + col);

    v8f c = {};
    #pragma unroll 4
    for (int k0 = 0; k0 < KK; k0 += 4) {
        const int ka = k0 + 2 * half;
        v2f a = { Arow[ka], Arow[ka + 1] };
        v2f b = { Bcol[(size_t)ka * HK], Bcol[(size_t)(ka + 1) * HK] };
        c = wmma4(a, b, c);
    }

    const float bias = bq[tn * 16 + col];
    #pragma unroll
    for (int j = 0; j < 8; ++j) {
        const int row = tm * 16 + j + 8 * half;
        qbuf[(size_t)row * HK + tn * 16 + col] = c[j] + bias;
    }
}

// ---------------------------------------------------------------------------
// Kernel 2: scores[b][h][l] = <keys[l,b,:] * rpe[l,b], q[b,h,:]>
// per wave: one batch b, one l-tile of 16.  A = mod_keys (16 x 256),
// B = q_b^T (256 x 16, cols 8..15 zero).   64*128 = 8192 waves
// keys is a 128MB single-pass stream: non-temporal loads + prefetch.
// ---------------------------------------------------------------------------
__global__ void k_scores(const float* __restrict__ keys,
                         const float* __restrict__ rpe,
                         const float* __restrict__ qbuf,
                         float* __restrict__ smat) {
    const int waveId = blockIdx.x * (blockDim.x >> 5) + (threadIdx.x >> 5);
    const int b    = waveId >> 7;     // /128 l-tiles
    const int lt   = waveId & 127;
    const int lane = threadIdx.x & 31;
    const int col  = lane & 15;
    const int half = lane >> 4;

    const int l_m = lt * 16 + col;    // this lane's A-row (l index)
    const float* keyRow = keys + ((size_t)l_m * BB + b) * KK;
    const float r = rpe[(size_t)l_m * BB + b];

    const bool bn = (col < HH);
    const float* qb = qbuf + (size_t)b * HK + (size_t)col * KK;

    // warm the stream: each lane's 1KB row = 8 cachelines
    __builtin_prefetch(keyRow, 0, 0);

    v8f c = {};
    #pragma unroll 8
    for (int k0 = 0; k0 < KK; k0 += 4) {
        const int ka = k0 + 2 * half;
        if ((k0 & 28) == 0)  // once per 32-float (128B) chunk: next line
            __builtin_prefetch(keyRow + k0 + 32, 0, 0);
        v2f kv = __builtin_nontemporal_load((const v2f*)(keyRow + ka));
        v2f a  = { kv.x * r, kv.y * r };
        v2f bb = { bn ? qb[ka] : 0.0f, bn ? qb[ka + 1] : 0.0f };
        c = wmma4(a, bb, c);
    }

    if (col < HH) {
        #pragma unroll
        for (int j = 0; j < 8; ++j) {
            const int l = lt * 16 + j + 8 * half;
            smat[((size_t)b * HH + col) * LL + l] = c[j];
        }
    }
}

// ---------------------------------------------------------------------------
// Kernel 3: in-place masked softmax over L per (b,h) row; weight=0 for l>=step
// one 256-thread block per (b,h) row; each thread holds 8 elems in registers
// ---------------------------------------------------------------------------
__global__ void k_softmax(float* __restrict__ s, const int* __restrict__ steps) {
    const int bh   = blockIdx.x;
    const int b    = bh >> 3;
    const int step = steps[b];
    float* row = s + (size_t)bh * LL;
    const int t = threadIdx.x;

    float v[8];
    float mx = -INFINITY;
    #pragma unroll
    for (int i = 0; i < 8; ++i) {
        const int l = t + i * 256;
        const float x = (l < step) ? row[l] : -INFINITY;
        v[i] = x;
        mx = fmaxf(mx, x);
    }

    __shared__ float red[256];
    red[t] = mx;
    __syncthreads();
    for (int off = 128; off > 0; off >>= 1) {
        if (t < off) red[t] = fmaxf(red[t], red[t + off]);
        __syncthreads();
    }
    mx = red[0];
    __syncthreads();

    float sum = 0.0f;
    #pragma unroll
    for (int i = 0; i < 8; ++i) {
        const int l = t + i * 256;
        const float e = (l < step) ? __expf(v[i] - mx) : 0.0f;
        v[i] = e;
        sum += e;
    }
    red[t] = sum;
    __syncthreads();
    for (int off = 128; off > 0; off >>= 1) {
        if (t < off) red[t] += red[t + off];
        __syncthreads();
    }
    const float inv = 1.0f / red[0];

    #pragma unroll
    for (int i = 0; i < 8; ++i) row[t + i * 256] = v[i] * inv;
}

// ---------------------------------------------------------------------------
// Kernel 4: read[b][h][v] = sum_l weight[b][h][l] * vals[l,b,v]
// per wave: one batch b, one 16-wide v-tile.  A = weight^T (8->16 x 2048),
// B = vals_b (2048 x 16).   64*16 = 1024 waves
// vals is a 128MB single-pass stream: non-temporal loads + row prefetch.
// ---------------------------------------------------------------------------
__global__ void k_read(const float* __restrict__ wmat,
                       const float* __restrict__ vals,
                       float* __restrict__ readbuf) {
    const int waveId = blockIdx.x * (blockDim.x >> 5) + (threadIdx.x >> 5);
    const int b    = waveId >> 4;
    const int tn   = waveId & 15;
    const int lane = threadIdx.x & 31;
    const int col  = lane & 15;
    const int half = lane >> 4;

    const bool am = (col < HH);
    const float* wrow  = wmat + ((size_t)b * HH + col) * LL;
    const float* vbase = vals + (size_t)b * VV + tn * 16 + col;

    v8f c = {};
    #pragma unroll 4
    for (int k0 = 0; k0 < LL; k0 += 4) {
        const int ka = k0 + 2 * half;
        if ((k0 & 12) == 0)  // once per 16 l-rows: prefetch 8 rows ahead
            __builtin_prefetch(vbase + (size_t)(k0 + 8) * (BB * VV), 0, 0);
        v2f a  = { am ? wrow[ka] : 0.0f, am ? wrow[ka + 1] : 0.0f };
        v2f bb = { __builtin_nontemporal_load(vbase + (size_t)ka * (BB * VV)),
                   __builtin_nontemporal_load(vbase + (size_t)(ka + 1) * (BB * VV)) };
        c = wmma4(a, bb, c);
    }

    if (half == 0) {  // valid D rows m = j (< 8)
        #pragma unroll
        for (int j = 0; j < 8; ++j)
            readbuf[((size_t)b * HH + j) * VV + tn * 16 + col] = c[j];
    }
}

// ---------------------------------------------------------------------------
// Kernel 5: out = read[B, HV] @ Wa[HV, V] + ba -> [B, V]
// tiles: tm in [0,4), tn in [0,16); 64 waves
// ---------------------------------------------------------------------------
__global__ void k_out(const float* __restrict__ readbuf,
                      const float* __restrict__ Wa,
                      const float* __restrict__ ba,
                      float* __restrict__ out) {
    const int waveId = blockIdx.x * (blockDim.x >> 5) + (threadIdx.x >> 5);
    const int tm   = waveId >> 4;
    const int tn   = waveId & 15;
    const int lane = threadIdx.x & 31;
    const int col  = lane & 15;
    const int half = lane >> 4;

    const float* Arow = readbuf + (size_t)(tm * 16 + col) * HV;
    const float* Bcol = Wa + tn * 16 + col;

    v8f c = {};
    #pragma unroll 2
    for (int k0 = 0; k0 < HV; k0 += 4) {
        const int ka = k0 + 2 * half;
        v2f a = { Arow[ka], Arow[ka + 1] };
        v2f b = { Bcol[(size_t)ka * VV], Bcol[(size_t)(ka + 1) * VV] };
        c = wmma4(a, b, c);
    }

    const float bias = ba[tn * 16 + col];
    #pragma unroll
    for (int j = 0; j < 8; ++j) {
        const int row = tm * 16 + j + 8 * half;
        out[(size_t)row * VV + tn * 16 + col] = c[j] + bias;
    }
}

// ---------------------------------------------------------------------------
extern "C" void kernel_launch(void* const* d_in, const int* in_sizes, int n_in,
                              void* d_out, int out_size, void* d_ws, size_t ws_size,
                              hipStream_t stream) {
    const float* query = (const float*)d_in[0];  // [B,K]
    const float* keys  = (const float*)d_in[1];  // [L,B,K]
    const float* vals  = (const float*)d_in[2];  // [L,B,V]
    const float* rpe   = (const float*)d_in[3];  // [L,B,1]
    const float* Wq    = (const float*)d_in[4];  // [K,HK]
    const float* bq    = (const float*)d_in[5];  // [HK]
    const float* Wa    = (const float*)d_in[6];  // [HV,V]
    const float* ba    = (const float*)d_in[7];  // [V]
    const int*   steps = (const int*)d_in[8];    // [B]
    float* out = (float*)d_out;                  // [B,V]

    float* ws = (float*)d_ws;
    float* qbuf    = ws;                               // B*HK   = 131072 f
    float* smat    = qbuf + (size_t)BB * HK;           // B*H*L  = 1048576 f
    float* readbuf = smat + (size_t)BB * HH * LL;      // B*H*V  = 131072 f

    // 1) q projection: 512 waves = 64 blocks x 8 waves
    k_qproj<<<64, 256, 0, stream>>>(query, Wq, bq, qbuf);
    // 2) scores: 8192 waves = 1024 blocks x 8 waves
    k_scores<<<1024, 256, 0, stream>>>(keys, rpe, qbuf, smat);
    // 3) softmax: one block per (b,h) row
    k_softmax<<<BB * HH, 256, 0, stream>>>(smat, steps);
    // 4) weighted value read: 1024 waves = 128 blocks x 8 waves
    k_read<<<128, 256, 0, stream>>>(smat, vals, readbuf);
    // 5) head aggregation: 64 waves = 8 blocks x 8 waves
    k_out<<<8, 256, 0, stream>>>(readbuf, Wa, ba, out);
}